// MtpDrafterANE_85839216378581
// MI455X (gfx1250) — compile-verified
//
#include <hip/hip_runtime.h>
#include <math.h>

typedef __attribute__((ext_vector_type(2))) float v2f;
typedef __attribute__((ext_vector_type(8))) float v8f;

#define H_      256
#define TH_     1536
#define FFN_    2048
#define VOCAB_  262144
#define NH_     4
#define EPS_    1e-6f
#define CAP_    30.0f

// ---------------- wave helpers (wave32) ----------------
__device__ __forceinline__ float waveSum(float v) {
#pragma unroll
  for (int o = 16; o > 0; o >>= 1) v += __shfl_xor(v, o, 32);
  return v;
}
__device__ __forceinline__ float waveMax(float v) {
#pragma unroll
  for (int o = 16; o > 0; o >>= 1) v = fmaxf(v, __shfl_xor(v, o, 32));
  return v;
}

// wave-cooperative dot of one row (n multiple of 4, 16B-aligned rows)
__device__ __forceinline__ float waveDot(const float* __restrict__ row,
                                         const float* __restrict__ vec, int n) {
  const float4* r4 = (const float4*)row;
  const float4* v4 = (const float4*)vec;
  int n4 = n >> 2;
  float s = 0.f;
  for (int i = (threadIdx.x & 31); i < n4; i += 32) {
    float4 a = r4[i]; float4 b = v4[i];
    s = fmaf(a.x, b.x, s); s = fmaf(a.y, b.y, s);
    s = fmaf(a.z, b.z, s); s = fmaf(a.w, b.w, s);
  }
  return waveSum(s);
}

__device__ __forceinline__ float gelu_exact(float x) {
  return 0.5f * x * (1.0f + erff(x * 0.70710678118654752440f));
}

// ---------------- generic kernels ----------------
// out[r] = dot(M[r,:], v); one wave per row; grid*blockDim/32 == rows
__global__ void mv_k(const float* __restrict__ M, const float* __restrict__ v,
                     float* __restrict__ out, int cols) {
  int wid = (blockIdx.x * blockDim.x + threadIdx.x) >> 5;
  float d = waveDot(M + (size_t)wid * cols, v, cols);
  if ((threadIdx.x & 31) == 0) out[wid] = d;
}

// x[r] = dot(in_e[r,:],emb) + dot(in_p[r,:],proj)
__global__ void embed_k(const float* __restrict__ ine, const float* __restrict__ inp,
                        const float* __restrict__ emb, const float* __restrict__ proj,
                        float* __restrict__ x) {
  int wid = (blockIdx.x * blockDim.x + threadIdx.x) >> 5;
  float s = waveDot(ine + (size_t)wid * TH_, emb, TH_) +
            waveDot(inp + (size_t)wid * TH_, proj, TH_);
  if ((threadIdx.x & 31) == 0) x[wid] = s;
}

// out[t] = x[t] * rsqrt(mean(x^2)+eps) * w[t]   (single block, blockDim==n)
__global__ void rmsnorm_k(const float* __restrict__ x, const float* __restrict__ w,
                          float* __restrict__ out, int n) {
  __shared__ float red[16]; __shared__ float sc;
  int t = threadIdx.x;
  float v = x[t];
  float s = waveSum(v * v);
  if ((t & 31) == 0) red[t >> 5] = s;
  __syncthreads();
  if (t == 0) {
    float tot = 0.f; int nw = blockDim.x >> 5;
    for (int j = 0; j < nw; ++j) tot += red[j];
    sc = rsqrtf(tot / (float)n + EPS_);
  }
  __syncthreads();
  out[t] = v * sc * w[t];
}

// x[t] += t1[t] * rsqrt(mean(t1^2)+eps) * w[t]
__global__ void rmsadd_k(float* __restrict__ x, const float* __restrict__ t1,
                         const float* __restrict__ w, int n) {
  __shared__ float red[16]; __shared__ float sc;
  int t = threadIdx.x;
  float v = t1[t];
  float s = waveSum(v * v);
  if ((t & 31) == 0) red[t >> 5] = s;
  __syncthreads();
  if (t == 0) {
    float tot = 0.f; int nw = blockDim.x >> 5;
    for (int j = 0; j < nw; ++j) tot += red[j];
    sc = rsqrtf(tot / (float)n + EPS_);
  }
  __syncthreads();
  x[t] += v * sc * w[t];
}

// per-head RMS norm with qn, then rope; grid=NH, blockDim=hd
__global__ void qnorm_rope_k(float* __restrict__ q, const float* __restrict__ qn,
                             const float* __restrict__ cosv, const float* __restrict__ sinv,
                             int hd) {
  __shared__ float sq[512]; __shared__ float red[16]; __shared__ float sc;
  int n = blockIdx.x, t = threadIdx.x;
  float v = q[n * hd + t];
  float s = waveSum(v * v);
  if ((t & 31) == 0) red[t >> 5] = s;
  __syncthreads();
  if (t == 0) {
    float tot = 0.f; int nw = blockDim.x >> 5;
    for (int j = 0; j < nw; ++j) tot += red[j];
    sc = rsqrtf(tot / (float)hd + EPS_);
  }
  __syncthreads();
  sq[t] = v * sc * qn[t];
  __syncthreads();
  int half = hd >> 1; float o;
  if (t < half) o = sq[t] * cosv[t] - sq[t + half] * sinv[t];
  else { int u = t - half; o = sq[t] * cosv[u] + sq[u] * sinv[u]; }
  q[n * hd + t] = o;
}

// attn[n*S+s] = dot(K[n,s,:], q[n,:]) + mask[s]; one wave per (n,s)
__global__ void scores_k(const float* __restrict__ K, const float* __restrict__ q,
                         const float* __restrict__ mask, float* __restrict__ attn,
                         int S, int hd) {
  int wid = (blockIdx.x * blockDim.x + threadIdx.x) >> 5;
  int n = wid / S, s = wid - n * S;
  float d = waveDot(K + (size_t)wid * hd, q + n * hd, hd);
  if ((threadIdx.x & 31) == 0) attn[wid] = d + mask[s];
}

// softmax over S per head; grid=NH, blockDim=256
__global__ void softmax_k(float* __restrict__ attn, int S) {
  float* a = attn + (size_t)blockIdx.x * S;
  __shared__ float red[8]; __shared__ float bc;
  int t = threadIdx.x;
  float m = -INFINITY;
  for (int i = t; i < S; i += 256) m = fmaxf(m, a[i]);
  m = waveMax(m);
  if ((t & 31) == 0) red[t >> 5] = m;
  __syncthreads();
  if (t == 0) {
    float mm = -INFINITY;
    for (int j = 0; j < 8; ++j) mm = fmaxf(mm, red[j]);
    bc = mm;
  }
  __syncthreads();
  float mm = bc, s = 0.f;
  for (int i = t; i < S; i += 256) { float e = expf(a[i] - mm); a[i] = e; s += e; }
  s = waveSum(s);
  __syncthreads();
  if ((t & 31) == 0) red[t >> 5] = s;
  __syncthreads();
  if (t == 0) {
    float tot = 0.f;
    for (int j = 0; j < 8; ++j) tot += red[j];
    bc = tot;
  }
  __syncthreads();
  float inv = 1.f / bc;
  for (int i = t; i < S; i += 256) a[i] *= inv;
}

// aout[n*hd+d] = dot(V[n,d,:], attn[n,:]) ; V layout (NH, hd, S); wave per (n,d)
__global__ void attnout_k(const float* __restrict__ Vv, const float* __restrict__ attn,
                          float* __restrict__ aout, int S, int hd) {
  int wid = (blockIdx.x * blockDim.x + threadIdx.x) >> 5;
  int n = wid / hd;
  float d = waveDot(Vv + (size_t)wid * S, attn + (size_t)n * S, S);
  if ((threadIdx.x & 31) == 0) aout[wid] = d;
}

// act[r] = gelu(g1w[r,:]·hv) * (g2w[r,:]·hv)
__global__ void ffngate_k(const float* __restrict__ g1w, const float* __restrict__ g2w,
                          const float* __restrict__ hv, float* __restrict__ act) {
  int wid = (blockIdx.x * blockDim.x + threadIdx.x) >> 5;
  float d1 = waveDot(g1w + (size_t)wid * H_, hv, H_);
  float d2 = waveDot(g2w + (size_t)wid * H_, hv, H_);
  if ((threadIdx.x & 31) == 0) act[wid] = gelu_exact(d1) * d2;
}

// ---------------- LM head: f32 WMMA matvec + softcap + per-block top-8 ----------------
// grid = VOCAB/128 blocks, 256 threads (8 waves), 16 rows per wave.
// A tile: lanes 0-15 hold row M=lane, K={k,k+1}; lanes 16-31 hold row M=lane-16, K={k+2,k+3}.
// B tile: broadcast h chunk across columns -> every D column equals the matvec.
__global__ void lmhead_k(const float* __restrict__ lm, const float* __restrict__ h,
                         float* __restrict__ cand_val, int* __restrict__ cand_idx) {
  __shared__ float sh[H_];
  __shared__ float slog[128];
  int t = threadIdx.x;
  sh[t] = h[t];
  __syncthreads();

  int lane = t & 31, wave = t >> 5;
  int rowBase = blockIdx.x * 128 + wave * 16;
  int m = lane & 15;
  int koff = (lane >> 4);                   // 0 or 1 (float2 units)
  const float2* row2 = (const float2*)(lm + (size_t)(rowBase + m) * H_);
  const float2* h2 = (const float2*)sh;

  v8f acc = {0.f, 0.f, 0.f, 0.f, 0.f, 0.f, 0.f, 0.f};
#pragma unroll 4
  for (int k = 0; k < H_; k += 4) {
    float2 ar = row2[(k >> 1) + koff];
    float2 br = h2[(k >> 1) + koff];
    v2f a; a[0] = ar.x; a[1] = ar.y;
    v2f b; b[0] = br.x; b[1] = br.y;
    acc = __builtin_amdgcn_wmma_f32_16x16x4_f32(
        /*neg_a=*/false, a, /*neg_b=*/false, b,
        /*c_mod=*/(short)0, acc, /*reuse_a=*/false, /*reuse_b=*/false);
  }

  // softcap and extract: lane 0 holds rows 0..7 (VGPR j -> M=j), lane 16 holds rows 8..15
  float capped[8];
#pragma unroll
  for (int j = 0; j < 8; ++j) capped[j] = tanhf(acc[j] * (1.0f / CAP_)) * CAP_;
  if (lane == 0) {
#pragma unroll
    for (int j = 0; j < 8; ++j) slog[wave * 16 + j] = capped[j];
  }
  if (lane == 16) {
#pragma unroll
    for (int j = 0; j < 8; ++j) slog[wave * 16 + 8 + j] = capped[j];
  }
  __syncthreads();

  if (t == 0) {
    for (int sel = 0; sel < 8; ++sel) {
      float best = -INFINITY; int bi = 0;
      for (int i = 0; i < 128; ++i)
        if (slog[i] > best) { best = slog[i]; bi = i; }   // first hit -> lowest index on tie
      cand_val[blockIdx.x * 8 + sel] = best;
      cand_idx[blockIdx.x * 8 + sel] = blockIdx.x * 128 + bi;
      slog[bi] = -INFINITY;
    }
  }
}

// single block: reduce ncand candidates to global top-8 (tie-break: lower vocab idx)
__global__ void topk_final_k(float* __restrict__ cv, const int* __restrict__ ci, int ncand,
                             int* __restrict__ out_ids, float* __restrict__ out_vals) {
  __shared__ float sv[256]; __shared__ int sx[256]; __shared__ int ss[256];
  int t = threadIdx.x;
  for (int iter = 0; iter < 8; ++iter) {
    float bv = -INFINITY; int bx = 0x7fffffff; int bs = -1;
    for (int i = t; i < ncand; i += 256) {
      float v = cv[i]; int x = ci[i];
      if (v > bv || (v == bv && x < bx)) { bv = v; bx = x; bs = i; }
    }
    sv[t] = bv; sx[t] = bx; ss[t] = bs;
    __syncthreads();
    for (int str = 128; str > 0; str >>= 1) {
      if (t < str) {
        float v2 = sv[t + str]; int x2 = sx[t + str];
        if (v2 > sv[t] || (v2 == sv[t] && x2 < sx[t])) {
          sv[t] = v2; sx[t] = x2; ss[t] = ss[t + str];
        }
      }
      __syncthreads();
    }
    if (t == 0) {
      out_ids[iter] = sx[0];
      out_vals[iter] = sv[0];
      if (ss[0] >= 0) cv[ss[0]] = -INFINITY;
    }
    __syncthreads();
  }
}

// ---------------- host orchestration ----------------
extern "C" void kernel_launch(void* const* d_in, const int* in_sizes, int n_in,
                              void* d_out, int out_size, void* d_ws, size_t ws_size,
                              hipStream_t stream) {
  (void)in_sizes; (void)n_in; (void)out_size; (void)ws_size;

  const float* embed_tok = (const float*)d_in[0];
  const float* proj_act  = (const float*)d_in[1];
  const float* kv13_k    = (const float*)d_in[2];
  const float* kv13_v    = (const float*)d_in[3];
  const float* kv14_k    = (const float*)d_in[4];
  const float* kv14_v    = (const float*)d_in[5];
  const float* cos_swa   = (const float*)d_in[6];
  const float* sin_swa   = (const float*)d_in[7];
  const float* cos_full  = (const float*)d_in[8];
  const float* sin_full  = (const float*)d_in[9];
  const float* mask_swa  = (const float*)d_in[10];
  const float* mask_full = (const float*)d_in[11];
  // params (insertion order): in_e, in_p, layers[0..3]{ln_in,ln_pa,ln_pf,ln_pff,qn,q_w,o_w,g1,g2,dn}, fn, lm, pp
  const float* in_e = (const float*)d_in[12];
  const float* in_p = (const float*)d_in[13];
  const float* fn_w = (const float*)d_in[14 + 4 * 10 + 0];
  const float* lm_w = (const float*)d_in[14 + 4 * 10 + 1];
  const float* pp_w = (const float*)d_in[14 + 4 * 10 + 2];

  float* ws = (float*)d_ws;
  float* X    = ws;                 // 256
  float* HV   = ws + 256;           // 256
  float* Q    = ws + 512;           // up to 2048
  float* ATT  = ws + 2560;          // up to 16384
  float* AOUT = ws + 18944;         // up to 2048
  float* T1   = ws + 20992;         // 256
  float* ACT  = ws + 21248;         // 2048
  float* HF   = ws + 23296;         // 256
  float* CV   = ws + 23552;         // 16384 candidate vals
  int*   CI   = (int*)(ws + 39936); // 16384 candidate idx

  float* out_f   = (float*)d_out;
  int*   out_ids = (int*)d_out;          // [0..7]
  float* out_val = out_f + 8;            // [8..15]
  float* out_pp  = out_f + 16;           // [16..1551]

  // x = in_e @ embed + in_p @ proj
  embed_k<<<32, 256, 0, stream>>>(in_e, in_p, embed_tok, proj_act, X);

  for (int i = 0; i < 4; ++i) {
    const int hd = (i == 3) ? 512 : 256;
    const int S  = (i == 3) ? 4096 : 512;
    const float* K  = (i == 3) ? kv14_k : kv13_k;
    const float* Vv = (i == 3) ? kv14_v : kv13_v;
    const float* cs = (i == 3) ? cos_full : cos_swa;
    const float* sn = (i == 3) ? sin_full : sin_swa;
    const float* mk = (i == 3) ? mask_full : mask_swa;

    const int base = 14 + 10 * i;
    const float* ln_in = (const float*)d_in[base + 0];
    const float* ln_pa = (const float*)d_in[base + 1];
    const float* ln_pf = (const float*)d_in[base + 2];
    const float* ln_pff= (const float*)d_in[base + 3];
    const float* qn    = (const float*)d_in[base + 4];
    const float* q_w   = (const float*)d_in[base + 5];
    const float* o_w   = (const float*)d_in[base + 6];
    const float* g1w   = (const float*)d_in[base + 7];
    const float* g2w   = (const float*)d_in[base + 8];
    const float* dnw   = (const float*)d_in[base + 9];

    // attention block
    rmsnorm_k<<<1, H_, 0, stream>>>(X, ln_in, HV, H_);
    mv_k<<<(NH_ * hd) / 8, 256, 0, stream>>>(q_w, HV, Q, H_);
    qnorm_rope_k<<<NH_, hd, 0, stream>>>(Q, qn, cs, sn, hd);
    scores_k<<<(NH_ * S) / 8, 256, 0, stream>>>(K, Q, mk, ATT, S, hd);
    softmax_k<<<NH_, 256, 0, stream>>>(ATT, S);
    attnout_k<<<(NH_ * hd) / 8, 256, 0, stream>>>(Vv, ATT, AOUT, S, hd);
    mv_k<<<H_ / 8, 256, 0, stream>>>(o_w, AOUT, T1, NH_ * hd);
    rmsadd_k<<<1, H_, 0, stream>>>(X, T1, ln_pa, H_);

    // FFN block
    rmsnorm_k<<<1, H_, 0, stream>>>(X, ln_pf, HV, H_);
    ffngate_k<<<FFN_ / 8, 256, 0, stream>>>(g1w, g2w, HV, ACT);
    mv_k<<<H_ / 8, 256, 0, stream>>>(dnw, ACT, T1, FFN_);
    rmsadd_k<<<1, H_, 0, stream>>>(X, T1, ln_pff, H_);
  }

  // final norm, LM head (WMMA) + top-k, projection
  rmsnorm_k<<<1, H_, 0, stream>>>(X, fn_w, HF, H_);
  lmhead_k<<<VOCAB_ / 128, 256, 0, stream>>>(lm_w, HF, CV, CI);
  topk_final_k<<<1, 256, 0, stream>>>(CV, CI, (VOCAB_ / 128) * 8, out_ids, out_val);
  mv_k<<<TH_ / 8, 256, 0, stream>>>(pp_w, HF, out_pp, H_);
}